// baseNeuron_4784593568441
// MI455X (gfx1250) — compile-verified
//
#include <hip/hip_runtime.h>
#include <math.h>

// LIF membrane scan over time axis.
//   x     : [B=32, C=64, T=32, H=28, W=28] f32
//   decay : [1] f32
//   out   : spikes in {0,1}, same shape as x
//
// Memory-bound: 411 MB total traffic -> ~18us at 23.3 TB/s.
// One workgroup per (b,c) plane: 224 threads, lanes tid<196 each own one
// float4 of the 28x28 plane across all 32 time steps. Time planes stream
// global->LDS via the gfx1250 async DMA path (ASYNCcnt), double-buffered in
// chunks of 8 steps. Because the async instruction offset is added to BOTH
// the LDS and global addresses, the LDS slot stride equals the plane stride
// (3136 B) and a block consumes exactly one plane per step -> no collisions,
// and every staged load is a single instruction (uniform sgpr base advanced
// per chunk, one shared vaddr = tid*16).

namespace {
constexpr int   T_STEPS  = 32;
constexpr int   PLANE4   = 196;               // 28*28/4 float4 per plane
constexpr int   PLANE_B  = PLANE4 * 16;       // 3136 bytes per time plane
constexpr int   PLANE_F  = PLANE4 * 4;        // 784 floats per time plane
constexpr int   BLOCK    = 224;               // 7 waves; tid<196 active
constexpr int   CHUNK    = 8;                 // time steps per async batch
constexpr int   NCHUNK   = T_STEPS / CHUNK;   // 4
constexpr float THRESH   = 0.5f;
}

// Async DMA global->LDS, GVS form: saddr(uniform) + vaddr(lane) + imm.
template <int OFF>
__device__ __forceinline__ void async_ld_b128(unsigned lds_addr, unsigned gvoff,
                                              const float* sbase) {
  asm volatile("global_load_async_to_lds_b128 %0, %1, %2 offset:%3"
               :: "v"(lds_addr), "v"(gvoff), "s"(sbase), "i"(OFF)
               : "memory");
}

template <int N>
__device__ __forceinline__ void wait_asynccnt() {
  asm volatile("s_wait_asynccnt %0" :: "i"(N) : "memory");
}

#define ISSUE_CHUNK(LDS, VOFF, SBASE)                              \
  do {                                                             \
    async_ld_b128<0 * PLANE_B>((LDS), (VOFF), (SBASE));            \
    async_ld_b128<1 * PLANE_B>((LDS), (VOFF), (SBASE));            \
    async_ld_b128<2 * PLANE_B>((LDS), (VOFF), (SBASE));            \
    async_ld_b128<3 * PLANE_B>((LDS), (VOFF), (SBASE));            \
    async_ld_b128<4 * PLANE_B>((LDS), (VOFF), (SBASE));            \
    async_ld_b128<5 * PLANE_B>((LDS), (VOFF), (SBASE));            \
    async_ld_b128<6 * PLANE_B>((LDS), (VOFF), (SBASE));            \
    async_ld_b128<7 * PLANE_B>((LDS), (VOFF), (SBASE));            \
  } while (0)

__global__ __launch_bounds__(BLOCK)
void lif_scan_async(const float* __restrict__ x,
                    const float* __restrict__ decay,
                    float* __restrict__ out) {
  // Double-buffered staging: 2 * 8 * 196 float4 = 50,176 B LDS.
  // Slot (buf,t) is exactly one 3136-B time plane; lane tid holds bytes
  // [tid*16, tid*16+16) of it -> conflict-free ds_load_b128 read-back.
  __shared__ float4 smem4[2][CHUNK][PLANE4];

  const int tid = threadIdx.x;
  if (tid >= PLANE4) return;                       // lanes 196..223 idle

  const unsigned bc = blockIdx.x;                  // (b*C + c) plane index

  // Uniform per-block bases (land in SGPRs); lane offset is just tid*16.
  const float* xplane = x   + (size_t)bc * (T_STEPS * PLANE_F);
  float*       oplane = out + (size_t)bc * (T_STEPS * PLANE_F);
  const unsigned gvoff = (unsigned)tid * 16u;

  const unsigned lds0 = (unsigned)(size_t)(&smem4[0][0][0]) + gvoff; // buffer 0
  const unsigned lds1 = (unsigned)(size_t)(&smem4[1][0][0]) + gvoff; // buffer 1

  // Prologue: stage chunk 0 through the async DMA engine.
  ISSUE_CHUNK(lds0, gvoff, xplane);

  const float d = 1.0f / (1.0f + expf(-decay[0])); // sigmoid(decay), uniform

  float4 m  = make_float4(0.f, 0.f, 0.f, 0.f);
  bool   px = false, py = false, pz = false, pw = false;

#pragma unroll
  for (int c = 0; c < NCHUNK; ++c) {
    const int cur = c & 1;

    if (c + 1 < NCHUNK) {
      // Issue chunk c+1 (scalar base advanced by 8 planes), then wait for
      // chunk c (in-order completion => asynccnt <= CHUNK).
      const unsigned nxt_lds = ((c + 1) & 1) ? lds1 : lds0;
      ISSUE_CHUNK(nxt_lds, gvoff, xplane + (c + 1) * (CHUNK * PLANE_F));
      wait_asynccnt<CHUNK>();
    } else {
      wait_asynccnt<0>();
    }

#pragma unroll
    for (int t = 0; t < CHUNK; ++t) {
      const int gt = c * CHUNK + t;                // compile-time constant
      const float4 xv = smem4[cur][t][tid];        // ds_load_b128

      if (gt == 0) {
        m = xv;                                    // init_mem = x[t=0]
      } else {
        // mem = spike ? x_t : fma(mem, d, x_t)    (spike in {0,1})
        m.x = px ? xv.x : fmaf(m.x, d, xv.x);
        m.y = py ? xv.y : fmaf(m.y, d, xv.y);
        m.z = pz ? xv.z : fmaf(m.z, d, xv.z);
        m.w = pw ? xv.w : fmaf(m.w, d, xv.w);
      }
      px = m.x > THRESH;
      py = m.y > THRESH;
      pz = m.z > THRESH;
      pw = m.w > THRESH;

      const float4 sv = make_float4(px ? 1.0f : 0.0f, py ? 1.0f : 0.0f,
                                    pz ? 1.0f : 0.0f, pw ? 1.0f : 0.0f);

      // uniform plane base + zext(tid*16) + imm -> SADDR global_store_b128
      *reinterpret_cast<float4*>(
          reinterpret_cast<unsigned char*>(oplane) + (size_t)gvoff
          + (size_t)(gt * PLANE_B)) = sv;
    }
  }
}

extern "C" void kernel_launch(void* const* d_in, const int* in_sizes, int n_in,
                              void* d_out, int out_size, void* d_ws, size_t ws_size,
                              hipStream_t stream) {
  (void)in_sizes; (void)n_in; (void)d_ws; (void)ws_size;

  const float* x     = (const float*)d_in[0];
  const float* decay = (const float*)d_in[1];
  float*       out   = (float*)d_out;

  // One block per (b,c) plane: out_size / (T*H*W) = 32*64 = 2048 blocks.
  const int n_planes = out_size / (T_STEPS * PLANE_F);   // 2048
  lif_scan_async<<<n_planes, BLOCK, 0, stream>>>(x, decay, out);
}